// VanillaVectorQuantizer_67362267070465
// MI455X (gfx1250) — compile-verified
//
#include <hip/hip_runtime.h>

typedef __attribute__((ext_vector_type(2))) float v2f;
typedef __attribute__((ext_vector_type(4))) float v4f;
typedef __attribute__((ext_vector_type(8))) float v8f;

#define D_DIM 64
#define K_CB  512
#define HW    4096                      // H*W
#define E_STRIDE 68                     // padded LDS row stride -> conflict-free b64 reads
#define LDS_E_FLOATS (K_CB * E_STRIDE)  // 34816 floats
#define LDS_BIAS_OFF LDS_E_FLOATS
#define LDS_IDX_OFF  (LDS_E_FLOATS + K_CB)
#define LDS_TOTAL_FLOATS (LDS_E_FLOATS + K_CB + 256)

__global__ __launch_bounds__(256)
void vq_argmin_wmma_kernel(const float* __restrict__ enc,   // [B, D, H, W]
                           const float* __restrict__ cb,    // [D, K]
                           float* __restrict__ out) {       // [B, D, H, W]
    extern __shared__ float smem[];
    float* eLds   = smem;                       // [K][E_STRIDE] codebook transposed [n][d]
    float* biasL  = smem + LDS_BIAS_OFF;        // [K] : -0.5*||e_n||^2
    int*   idxLds = (int*)(smem + LDS_IDX_OFF); // [8 waves][32 rows]

    const int tid  = threadIdx.x;
    const int wave = tid >> 5;
    const int lane = tid & 31;
    const int m    = lane & 15;
    const int half = lane >> 4;        // 0: lanes 0-15, 1: lanes 16-31

    // 32 consecutive flattened rows per wave (32 | 4096 -> single batch image per wave)
    const int rowBase = (blockIdx.x * 8 + wave) * 32;
    const int b       = rowBase >> 12;           // / HW
    const int pBase   = rowBase & (HW - 1);

    // ---- Issue A-fragment global loads FIRST (overlap latency with LDS staging).
    //      32-bit 16x4 A layout: lane L, vgpr r <-> (M = L%16, K = 4t + 2*(L/16) + r).
    //      Two row tiles per wave: rows pBase+m and pBase+16+m. Non-temporal: enc
    //      is streamed exactly once -> don't rinse L2 (keep it for the codebook). ----
    const float* xptr0 = enc + (size_t)b * (D_DIM * HW) + (pBase + m);
    const float* xptr1 = xptr0 + 16;
    v2f a0[16], a1[16];
    #pragma unroll
    for (int t = 0; t < 16; ++t) {
        const size_t kk = (size_t)(4 * t + 2 * half) * HW;
        a0[t].x = __builtin_nontemporal_load(xptr0 + kk);
        a0[t].y = __builtin_nontemporal_load(xptr0 + kk + HW);
        a1[t].x = __builtin_nontemporal_load(xptr1 + kk);
        a1[t].y = __builtin_nontemporal_load(xptr1 + kk + HW);
    }

    // ---- Stage codebook into LDS, transposed to [n][d] (coalesced b128 global reads) ----
    #pragma unroll 4
    for (int i = 0; i < 32; ++i) {
        int flat = (i * 256 + tid) * 4;         // float index into cb[D*K]
        int d = flat >> 9;                      // /512
        int n = flat & 511;
        v4f v = *(const v4f*)(cb + flat);
        eLds[(n + 0) * E_STRIDE + d] = v.x;
        eLds[(n + 1) * E_STRIDE + d] = v.y;
        eLds[(n + 2) * E_STRIDE + d] = v.z;
        eLds[(n + 3) * E_STRIDE + d] = v.w;
    }
    __syncthreads();

    // ---- Bias: -0.5 * ||e_n||^2 ----
    for (int n = tid; n < K_CB; n += 256) {
        const float* row = eLds + n * E_STRIDE;
        float s = 0.f;
        #pragma unroll
        for (int d = 0; d < D_DIM; d += 2) {
            v2f v = *(const v2f*)(row + d);
            s += v.x * v.x + v.y * v.y;
        }
        biasL[n] = -0.5f * s;
    }
    __syncthreads();

    float bestA[8], bestB[8];
    int   bidxA[8], bidxB[8];
    #pragma unroll
    for (int r = 0; r < 8; ++r) {
        bestA[r] = -3.4e38f; bidxA[r] = 0;
        bestB[r] = -3.4e38f; bidxB[r] = 0;
    }

    // ---- Sweep 32 column tiles; each B fragment (ds_load_2addr_b64) feeds TWO wmmas ----
    for (int tile = 0; tile < 32; ++tile) {
        const int ncol = tile * 16 + m;                    // this lane's codebook column
        const float* erow = eLds + ncol * E_STRIDE + 2 * half;
        v8f c0 = {}, c1 = {};
        #pragma unroll
        for (int t = 0; t < 16; ++t) {
            v2f bf = *(const v2f*)(erow + 4 * t);          // conflict-free b64
            c0 = __builtin_amdgcn_wmma_f32_16x16x4_f32(
                     false, a0[t], false, bf, (short)0, c0, false, false);
            c1 = __builtin_amdgcn_wmma_f32_16x16x4_f32(
                     false, a1[t], false, bf, (short)0, c1, false, false);
        }
        const float nb = biasL[ncol];                      // -0.5||e||^2
        #pragma unroll
        for (int r = 0; r < 8; ++r) {
            float s0 = c0[r] + nb;
            if (s0 > bestA[r]) { bestA[r] = s0; bidxA[r] = ncol; }
            float s1 = c1[r] + nb;
            if (s1 > bestB[r]) { bestB[r] = s1; bidxB[r] = ncol; }
        }
    }

    // ---- Cross-lane argmax within each 16-lane half (xor<=8 stays inside the half);
    //      tie-break toward the smaller index to match argmin-first semantics ----
    #pragma unroll
    for (int sh = 1; sh <= 8; sh <<= 1) {
        #pragma unroll
        for (int r = 0; r < 8; ++r) {
            float oa = __shfl_xor(bestA[r], sh, 32);
            int   ia = __shfl_xor(bidxA[r], sh, 32);
            if (oa > bestA[r] || (oa == bestA[r] && ia < bidxA[r])) {
                bestA[r] = oa; bidxA[r] = ia;
            }
            float ob = __shfl_xor(bestB[r], sh, 32);
            int   ib = __shfl_xor(bidxB[r], sh, 32);
            if (ob > bestB[r] || (ob == bestB[r] && ib < bidxB[r])) {
                bestB[r] = ob; bidxB[r] = ib;
            }
        }
    }

    // half 0 holds rows 0..7 (vgpr r -> M=r); half 1 holds rows 8..15 (M=r+8)
    if (m == 0) {
        #pragma unroll
        for (int r = 0; r < 8; ++r) {
            idxLds[wave * 32 + half * 8 + r]      = bidxA[r];
            idxLds[wave * 32 + 16 + half * 8 + r] = bidxB[r];
        }
    }
    __syncthreads();

    // ---- Gather + write: lane = row-in-tile => 128B coalesced runs per d.
    //      Output is written exactly once -> non-temporal stores. ----
    const int   myIdx = idxLds[wave * 32 + lane];
    float*      obase = out + (size_t)b * (D_DIM * HW) + (pBase + lane);
    const float* crow = eLds + myIdx * E_STRIDE;
    #pragma unroll
    for (int d = 0; d < D_DIM; ++d) {
        __builtin_nontemporal_store(crow[d], obase + (size_t)d * HW);
    }
}

extern "C" void kernel_launch(void* const* d_in, const int* in_sizes, int n_in,
                              void* d_out, int out_size, void* d_ws, size_t ws_size,
                              hipStream_t stream) {
    const float* enc = (const float*)d_in[0];   // [32, 64, 64, 64] f32
    const float* cb  = (const float*)d_in[1];   // [64, 512] f32
    float* out       = (float*)d_out;           // [32, 64, 64, 64] f32
    (void)in_sizes; (void)n_in; (void)out_size; (void)d_ws; (void)ws_size;

    const size_t ldsBytes = LDS_TOTAL_FLOATS * sizeof(float);  // ~139 KB (< 320 KB WGP LDS)
    // 131072 rows / (8 waves * 32 rows) = 512 workgroups
    vq_argmin_wmma_kernel<<<512, 256, ldsBytes, stream>>>(enc, cb, out);
}